// GraphConvBlock_7945689498011
// MI455X (gfx1250) — compile-verified
//
#include <hip/hip_runtime.h>
#include <hip/hip_bf16.h>
#include <float.h>

// ---------------- problem constants (match reference) ----------------
#define NN   50000
#define EE   400000
#define F_IN 128
#define NH   4
#define CC   32
#define HC   128   // NH*CC
#define ED   16
#define BN_EPS 1e-5f

typedef float v2f __attribute__((ext_vector_type(2)));
typedef float v8f __attribute__((ext_vector_type(8)));

// ---------------------------------------------------------------------
// WMMA fp32 GEMM: out[M,128] = A[M,128] @ W[128,128] + bias
// grid.x = M/16 row tiles; block = 256 threads = 8 waves; wave w owns the
// 16-wide column tile w. Uses V_WMMA_F32_16X16X4_F32, K stepped by 4.
// A-frag layout (16x4 f32): lanes 0-15 hold K={0,1}, lanes 16-31 K={2,3}.
// C/D layout: VGPR r -> M=r (lanes 0-15) / M=r+8 (lanes 16-31), N=lane&15.
// ---------------------------------------------------------------------
__global__ __launch_bounds__(256) void gemm128_wmma(
    const float* __restrict__ A, const float* __restrict__ W,
    const float* __restrict__ bias, float* __restrict__ out)
{
    const int rb   = blockIdx.x;
    const int wave = threadIdx.x >> 5;      // 0..7 -> column tile
    const int lane = threadIdx.x & 31;
    const int m    = lane & 15;
    const int kk   = (lane >> 4) * 2;       // 0 or 2
    const int col  = wave * 16 + m;
    const float* Arow = A + (size_t)(rb * 16 + m) * 128;

    v8f c = {};
    #pragma unroll 8
    for (int k0 = 0; k0 < 128; k0 += 4) {
        v2f a, b;
        a.x = Arow[k0 + kk];
        a.y = Arow[k0 + kk + 1];
        b.x = W[(size_t)(k0 + kk)     * HC + col];
        b.y = W[(size_t)(k0 + kk + 1) * HC + col];
        c = __builtin_amdgcn_wmma_f32_16x16x4_f32(
                false, a, false, b, (short)0, c, false, false);
    }

    const float bc = bias[col];
    const int rbase = rb * 16 + ((lane >> 4) ? 8 : 0);
    #pragma unroll
    for (int r = 0; r < 8; ++r)
        out[(size_t)(rbase + r) * HC + col] = c[r] + bc;
}

// ---------------- float atomic max via int/uint ordering trick --------
__device__ __forceinline__ void atomicMaxF(float* addr, float val)
{
    if (val >= 0.0f) atomicMax((int*)addr, __float_as_int(val));
    else             atomicMin((unsigned int*)addr, __float_as_uint(val));
}

__global__ void init_nodehead(float* amax, float* den, int n)
{
    int t = blockIdx.x * blockDim.x + threadIdx.x;
    if (t < n) { amax[t] = -FLT_MAX; den[t] = 0.0f; }
}

// ---------------- pass A: raw scores + segment max --------------------
__global__ void attn_score(const int* __restrict__ src, const int* __restrict__ dst,
                           const float* __restrict__ q, const float* __restrict__ k,
                           const float* __restrict__ ea, const float* __restrict__ ew,
                           float* __restrict__ araw, float* __restrict__ amax)
{
    int t = blockIdx.x * blockDim.x + threadIdx.x;
    if (t >= EE * NH) return;
    const int e = t >> 2, h = t & 3;
    const int s = src[e], d = dst[e];

    float w[ED];
    #pragma unroll
    for (int dd = 0; dd < ED; ++dd) w[dd] = ea[(size_t)e * ED + dd];

    const float* qr = q + (size_t)d * HC + h * CC;
    const float* kr = k + (size_t)s * HC + h * CC;
    const float* ewh = ew + h * CC;

    float acc = 0.0f;
    #pragma unroll 4
    for (int c = 0; c < CC; ++c) {
        float emb = 0.0f;
        #pragma unroll
        for (int dd = 0; dd < ED; ++dd) emb += w[dd] * ewh[dd * HC + c];
        acc += qr[c] * (kr[c] + emb);
    }
    acc *= 0.17677669529663687f;   // 1/sqrt(32)
    araw[t] = acc;
    atomicMaxF(&amax[d * NH + h], acc);
}

// ---------------- pass B: exp + segment sum ---------------------------
__global__ void attn_expden(const int* __restrict__ dst,
                            const float* __restrict__ amax,
                            float* __restrict__ p, float* __restrict__ den)
{
    int t = blockIdx.x * blockDim.x + threadIdx.x;
    if (t >= EE * NH) return;
    const int e = t >> 2, h = t & 3;
    const int d = dst[e];
    float v = __expf(p[t] - amax[d * NH + h]);
    p[t] = v;
    atomicAdd(&den[d * NH + h], v);
}

// ---------------- pass C: weighted message scatter --------------------
__global__ void scatter_msg(const int* __restrict__ src, const int* __restrict__ dst,
                            const float* __restrict__ vv, const float* __restrict__ ea,
                            const float* __restrict__ ew, const float* __restrict__ p,
                            const float* __restrict__ den, float* __restrict__ acc)
{
    int t = blockIdx.x * blockDim.x + threadIdx.x;
    if (t >= EE * NH) return;
    const int e = t >> 2, h = t & 3;
    const int s = src[e], d = dst[e];

    float dn = den[d * NH + h];
    float wgt = p[t] / (dn > 0.0f ? dn : 1.0f);

    float w[ED];
    #pragma unroll
    for (int dd = 0; dd < ED; ++dd) w[dd] = ea[(size_t)e * ED + dd];

    const float* vr  = vv + (size_t)s * HC + h * CC;
    const float* ewh = ew + h * CC;
    float* ar = acc + (size_t)d * HC + h * CC;

    #pragma unroll 4
    for (int c = 0; c < CC; ++c) {
        float emb = 0.0f;
        #pragma unroll
        for (int dd = 0; dd < ED; ++dd) emb += w[dd] * ewh[dd * HC + c];
        atomicAdd(&ar[c], (vr[c] + emb) * wgt);
    }
}

// ---------------- BN: per-channel stats -> scale/shift ----------------
__global__ __launch_bounds__(256) void bn_stats(
    const float* __restrict__ x, const float* __restrict__ g,
    const float* __restrict__ b, float* __restrict__ scale,
    float* __restrict__ shift)
{
    const int ch = blockIdx.x;          // 0..127
    __shared__ float ssum[256], ssq[256];
    float s = 0.0f, s2 = 0.0f;
    for (int i = threadIdx.x; i < NN; i += 256) {
        float v = x[(size_t)i * HC + ch];
        s += v; s2 += v * v;
    }
    ssum[threadIdx.x] = s; ssq[threadIdx.x] = s2;
    __syncthreads();
    for (int off = 128; off > 0; off >>= 1) {
        if (threadIdx.x < off) {
            ssum[threadIdx.x] += ssum[threadIdx.x + off];
            ssq [threadIdx.x] += ssq [threadIdx.x + off];
        }
        __syncthreads();
    }
    if (threadIdx.x == 0) {
        float mean = ssum[0] / (float)NN;
        float var  = ssq[0] / (float)NN - mean * mean;   // biased, matches jnp.var
        float inv  = rsqrtf(var + BN_EPS);
        float sc   = g[ch] * inv;
        scale[ch] = sc;
        shift[ch] = b[ch] - mean * sc;
    }
}

__global__ void bn_apply_relu(const float* __restrict__ x,
                              const float* __restrict__ scale,
                              const float* __restrict__ shift,
                              float* __restrict__ y)
{
    int t = blockIdx.x * blockDim.x + threadIdx.x;
    if (t >= NN * HC) return;
    int ch = t & (HC - 1);
    float v = x[t] * scale[ch] + shift[ch];
    y[t] = v > 0.0f ? v : 0.0f;
}

// ---------------------------------------------------------------------
static void run_layer(const float* xin, const int* src, const int* dst,
                      const float* ea,
                      const float* qw, const float* qb,
                      const float* kw, const float* kb,
                      const float* vw, const float* vb,
                      const float* ew,
                      const float* sw, const float* sb,
                      const float* g,  const float* b,
                      float* xout,
                      float* q, float* k, float* v, float* acc,
                      float* p, float* amax, float* den,
                      float* scale, float* shift,
                      hipStream_t stream)
{
    const int mtiles = NN / 16;       // 3125
    gemm128_wmma<<<mtiles, 256, 0, stream>>>(xin, qw, qb, q);
    gemm128_wmma<<<mtiles, 256, 0, stream>>>(xin, kw, kb, k);
    gemm128_wmma<<<mtiles, 256, 0, stream>>>(xin, vw, vb, v);
    gemm128_wmma<<<mtiles, 256, 0, stream>>>(xin, sw, sb, acc);   // skip init

    const int nh = NN * NH;
    init_nodehead<<<(nh + 255) / 256, 256, 0, stream>>>(amax, den, nh);

    const int ethreads = EE * NH;
    const int eblocks  = (ethreads + 255) / 256;
    attn_score <<<eblocks, 256, 0, stream>>>(src, dst, q, k, ea, ew, p, amax);
    attn_expden<<<eblocks, 256, 0, stream>>>(dst, amax, p, den);
    scatter_msg<<<eblocks, 256, 0, stream>>>(src, dst, v, ea, ew, p, den, acc);

    bn_stats<<<HC, 256, 0, stream>>>(acc, g, b, scale, shift);
    bn_apply_relu<<<(NN * HC + 255) / 256, 256, 0, stream>>>(acc, scale, shift, xout);
}

extern "C" void kernel_launch(void* const* d_in, const int* in_sizes, int n_in,
                              void* d_out, int out_size, void* d_ws, size_t ws_size,
                              hipStream_t stream)
{
    (void)in_sizes; (void)n_in; (void)out_size; (void)ws_size;

    const float* x  = (const float*)d_in[0];
    const int*   ei = (const int*)  d_in[1];
    const float* ea = (const float*)d_in[2];
    const int* src = ei;            // edge_index[0]
    const int* dst = ei + EE;       // edge_index[1]

    // layer params: q_w,q_b,k_w,k_b,v_w,v_b,e_w,s_w,s_b,bn_g,bn_b  (11 each)
    const float* P1[11]; const float* P2[11];
    for (int i = 0; i < 11; ++i) { P1[i] = (const float*)d_in[3 + i];
                                   P2[i] = (const float*)d_in[14 + i]; }

    // workspace carve-up (floats)
    float* ws = (float*)d_ws;
    size_t off = 0;
    float* q    = ws + off; off += (size_t)NN * HC;
    float* k    = ws + off; off += (size_t)NN * HC;
    float* v    = ws + off; off += (size_t)NN * HC;
    float* acc  = ws + off; off += (size_t)NN * HC;
    float* p    = ws + off; off += (size_t)EE * NH;
    float* amax = ws + off; off += (size_t)NN * NH;
    float* den  = ws + off; off += (size_t)NN * NH;
    float* scale = ws + off; off += HC;
    float* shift = ws + off; off += HC;

    float* x1 = (float*)d_out;                      // first output
    float* x2 = (float*)d_out + (size_t)NN * HC;    // second output

    run_layer(x,  src, dst, ea,
              P1[0], P1[1], P1[2], P1[3], P1[4], P1[5], P1[6], P1[7], P1[8],
              P1[9], P1[10], x1,
              q, k, v, acc, p, amax, den, scale, shift, stream);

    run_layer(x1, src, dst, ea,
              P2[0], P2[1], P2[2], P2[3], P2[4], P2[5], P2[6], P2[7], P2[8],
              P2[9], P2[10], x2,
              q, k, v, acc, p, amax, den, scale, shift, stream);
}